// FocalLoss_54752243089617
// MI455X (gfx1250) — compile-verified
//
#include <hip/hip_runtime.h>
#include <math.h>

// ---------------- constants from the reference ----------------
#define ALPHA_F      0.25f
#define HUBER_DELTA  0.5f
#define REG_BETA     (1.0f / 9.0f)
#define TPB          256
#define MAX_ANN      64

typedef __attribute__((ext_vector_type(4))) float f4;
typedef __attribute__((ext_vector_type(2))) float v2f;
typedef __attribute__((ext_vector_type(8))) float v8f;

// ---------------------------------------------------------------
// Cross-lane (wave32) sum via V_WMMA_F32_16X16X4_F32 with B = ones.
// D[m][n] = sum_k A[m][k].  A vgpr0: lane l<16 -> A[l][0], lane l>=16 ->
// A[l-16][2] (ISA 7.12.2 "32-bit A-Matrix 16x4").  Pass 1 produces row
// sums; summing the 8 D vgprs per lane gives S_low (lanes 0-15) and
// S_high (lanes 16-31); pass 2 computes D[m][n] = s[m] + s[m+16] =
// S_low + S_high = total, broadcast to every lane.  Exact f32 sum.
// Must be called with full EXEC (all threads of the block, no divergence).
// ---------------------------------------------------------------
__device__ __forceinline__ float wave_reduce_wmma(float v) {
  v2f a;   a.x = v;    a.y = 0.0f;
  v2f one; one.x = 1.0f; one.y = 1.0f;
  v8f c = {};
  v8f d = __builtin_amdgcn_wmma_f32_16x16x4_f32(
      false, a, false, one, (short)0, c, false, false);
  float s = d[0] + d[1] + d[2] + d[3] + d[4] + d[5] + d[6] + d[7];
  v2f a2; a2.x = s; a2.y = 0.0f;
  v8f d2 = __builtin_amdgcn_wmma_f32_16x16x4_f32(
      false, a2, false, one, (short)0, c, false, false);
  return d2[0];
}

__device__ __forceinline__ float smooth_l1(float diff) {
  float ad = fabsf(diff);
  return (ad <= REG_BETA) ? (0.5f * 9.0f * ad * ad) : (ad - 0.5f / 9.0f);
}

// ---------------------------------------------------------------
// Kernel 1: one thread per anchor. IoU argmax over M annotations,
// focal cls loss over C classes, smooth-L1 + Huber on positives.
// Block-reduces {cls_sum, pos_cnt, reg_sum, dist_sum} -> partials.
// ---------------------------------------------------------------
__global__ __launch_bounds__(TPB) void focal_per_anchor_kernel(
    const float* __restrict__ cls, const float* __restrict__ reg,
    const float* __restrict__ dist, const float* __restrict__ anchors,
    const float* __restrict__ ann, float* __restrict__ partial,
    int A, int C, int M, int nBlocks)
{
  const int b   = blockIdx.y;
  const int a   = blockIdx.x * TPB + threadIdx.x;
  const int tid = threadIdx.x;
  if (M > MAX_ANN) M = MAX_ANN;

  __shared__ float sann[MAX_ANN * 6];
  for (int i = tid; i < M * 6; i += TPB)
    sann[i] = ann[(size_t)b * (size_t)M * 6 + i];
  __syncthreads();

  float cls_sum = 0.0f, pos_cnt = 0.0f, reg_sum = 0.0f, dist_sum = 0.0f;

  if (a < A) {
    const float ax1 = anchors[a * 4 + 0], ay1 = anchors[a * 4 + 1];
    const float ax2 = anchors[a * 4 + 2], ay2 = anchors[a * 4 + 3];
    const float aw = ax2 - ax1, ah = ay2 - ay1;
    const float acx = ax1 + 0.5f * aw, acy = ay1 + 0.5f * ah;
    const float aarea = aw * ah;

    // IoU argmax (strict > replicates jnp.argmax first-occurrence)
    float best = -1.0f; int arg = 0;
    for (int m = 0; m < M; ++m) {
      const float* g = &sann[m * 6];
      float iou = -1.0f;
      if (g[4] != -1.0f) {
        float iw = fmaxf(fminf(ax2, g[2]) - fmaxf(ax1, g[0]), 0.0f);
        float ih = fmaxf(fminf(ay2, g[3]) - fmaxf(ay1, g[1]), 0.0f);
        float inter = iw * ih;
        float barea = (g[2] - g[0]) * (g[3] - g[1]);
        float ua = fmaxf(aarea + barea - inter, 1e-8f);
        iou = inter / ua;
      }
      if (iou > best) { best = iou; arg = m; }
    }
    const bool pos = best >= 0.5f;
    const bool neg = best < 0.4f;   // else: ignored anchor, contributes 0
    const float* g = &sann[arg * 6];
    const int label = (int)g[4];

    if (pos || neg) {
      // streaming read of this anchor's C classes: NT B128 loads
      const f4* cp = (const f4*)(cls + ((size_t)b * A + a) * (size_t)C);
      const int c4 = C >> 2;
      for (int i = 0; i < c4; ++i) {
        f4 x4 = __builtin_nontemporal_load(cp + i);
#pragma unroll
        for (int j = 0; j < 4; ++j) {
          float x = fminf(fmaxf(x4[j], 1e-4f), 1.0f - 1e-4f);
          const int cidx = i * 4 + j;
          float term;
          if (pos && cidx == label) {
            float om = 1.0f - x;
            term = ALPHA_F * om * om * (-__logf(x));
          } else {
            term = (1.0f - ALPHA_F) * x * x * (-__logf(1.0f - x));
          }
          cls_sum += term;
        }
      }
      for (int cidx = c4 * 4; cidx < C; ++cidx) {   // generic tail
        float x = cls[((size_t)b * A + a) * (size_t)C + cidx];
        x = fminf(fmaxf(x, 1e-4f), 1.0f - 1e-4f);
        if (pos && cidx == label) {
          float om = 1.0f - x;
          cls_sum += ALPHA_F * om * om * (-__logf(x));
        } else {
          cls_sum += (1.0f - ALPHA_F) * x * x * (-__logf(1.0f - x));
        }
      }
    }

    if (pos) {
      pos_cnt = 1.0f;
      const float gw_raw = g[2] - g[0], gh_raw = g[3] - g[1];
      const float gcx = g[0] + 0.5f * gw_raw, gcy = g[1] + 0.5f * gh_raw;
      const float gw = fmaxf(gw_raw, 1.0f), gh = fmaxf(gh_raw, 1.0f);
      const float t0 = ((gcx - acx) / aw) / 0.1f;
      const float t1 = ((gcy - acy) / ah) / 0.1f;
      const float t2 = logf(gw / aw) / 0.2f;
      const float t3 = logf(gh / ah) / 0.2f;
      const float* rp = reg + ((size_t)b * A + a) * 4;
      reg_sum = smooth_l1(t0 - rp[0]) + smooth_l1(t1 - rp[1]) +
                smooth_l1(t2 - rp[2]) + smooth_l1(t3 - rp[3]);
      const float dd = fabsf(dist[(size_t)b * A + a] - g[5]);
      dist_sum = (dd <= HUBER_DELTA) ? 0.5f * dd * dd
                                     : HUBER_DELTA * (dd - 0.5f * HUBER_DELTA);
    }
  }

  // deterministic block tree reduction of 4 quantities
  __shared__ float sred[4][TPB];
  sred[0][tid] = cls_sum; sred[1][tid] = pos_cnt;
  sred[2][tid] = reg_sum; sred[3][tid] = dist_sum;
  __syncthreads();
  for (int s = TPB / 2; s > 0; s >>= 1) {
    if (tid < s) {
      sred[0][tid] += sred[0][tid + s];
      sred[1][tid] += sred[1][tid + s];
      sred[2][tid] += sred[2][tid + s];
      sred[3][tid] += sred[3][tid + s];
    }
    __syncthreads();
  }
  if (tid == 0) {
    float* p = partial + ((size_t)b * nBlocks + blockIdx.x) * 4;
    p[0] = sred[0][0]; p[1] = sred[1][0];
    p[2] = sred[2][0]; p[3] = sred[3][0];
  }
}

// ---------------------------------------------------------------
// Kernel 2: per-sample reduction of per-block partials. Final 32-lane
// sum goes through the WMMA ones-matrix reducer (exercises v_wmma on
// gfx1250). Applies the num_pos / denom logic.
// ---------------------------------------------------------------
__global__ __launch_bounds__(TPB) void focal_finalize_kernel(
    const float* __restrict__ partial, float* __restrict__ sampleLoss,
    int nBlocks)
{
  const int b = blockIdx.x;
  const int tid = threadIdx.x;
  float loc[4] = {0.0f, 0.0f, 0.0f, 0.0f};
  const float* base = partial + (size_t)b * nBlocks * 4;
  for (int i = tid; i < nBlocks; i += TPB) {
    loc[0] += base[i * 4 + 0]; loc[1] += base[i * 4 + 1];
    loc[2] += base[i * 4 + 2]; loc[3] += base[i * 4 + 3];
  }
  __shared__ float sred[4][TPB];
#pragma unroll
  for (int q = 0; q < 4; ++q) sred[q][tid] = loc[q];
  __syncthreads();
  for (int s = TPB / 2; s >= 32; s >>= 1) {
    if (tid < s) {
#pragma unroll
      for (int q = 0; q < 4; ++q) sred[q][tid] += sred[q][tid + s];
    }
    __syncthreads();
  }
  // 32 partials per quantity left; every thread (full EXEC in each wave)
  // runs the WMMA reducer -> total broadcast to all lanes.
  const int lane = tid & 31;
  float tot[4];
#pragma unroll
  for (int q = 0; q < 4; ++q) tot[q] = wave_reduce_wmma(sred[q][lane]);

  if (tid == 0) {
    const float np = tot[1];
    const float denom = fmaxf(np, 1.0f);
    sampleLoss[b * 3 + 0] = tot[0] / denom;
    sampleLoss[b * 3 + 1] = (np > 0.0f) ? tot[2] / (denom * 4.0f) : 0.0f;
    sampleLoss[b * 3 + 2] = (np > 0.0f) ? tot[3] / denom : 0.0f;
  }
}

// Kernel 3: batch means -> d_out[0..2]
__global__ void focal_mean_kernel(const float* __restrict__ sampleLoss,
                                  float* __restrict__ out, int B)
{
  const int q = threadIdx.x;
  if (q < 3) {
    float s = 0.0f;
    for (int b = 0; b < B; ++b) s += sampleLoss[b * 3 + q];
    out[q] = s / (float)B;
  }
}

extern "C" void kernel_launch(void* const* d_in, const int* in_sizes, int n_in,
                              void* d_out, int out_size, void* d_ws, size_t ws_size,
                              hipStream_t stream)
{
  const float* cls     = (const float*)d_in[0];
  const float* reg     = (const float*)d_in[1];
  const float* dist    = (const float*)d_in[2];
  const float* anchors = (const float*)d_in[3];
  const float* ann     = (const float*)d_in[4];

  const int A = in_sizes[3] / 4;            // anchors: [1, A, 4]
  const int B = in_sizes[2] / A;            // distances: [B, A, 1]
  const int C = in_sizes[0] / (B * A);      // classifications: [B, A, C]
  const int M = in_sizes[4] / (B * 6);      // annotations: [B, M, 6]

  const int nBlocks = (A + TPB - 1) / TPB;

  float* partial    = (float*)d_ws;                       // B*nBlocks*4 floats
  float* sampleLoss = partial + (size_t)B * nBlocks * 4;  // B*3 floats

  dim3 g1(nBlocks, B);
  focal_per_anchor_kernel<<<g1, TPB, 0, stream>>>(
      cls, reg, dist, anchors, ann, partial, A, C, M, nBlocks);
  focal_finalize_kernel<<<B, TPB, 0, stream>>>(partial, sampleLoss, nBlocks);
  focal_mean_kernel<<<1, 64, 0, stream>>>(sampleLoss, (float*)d_out, B);
}